// CollapseEngine_19567871000775
// MI455X (gfx1250) — compile-verified
//
#include <hip/hip_runtime.h>

// CollapseEngine on MI455X (gfx1250): single fused kernel.
// - h tile resident in LDS across all 6 layers (compute-bound problem).
// - MLP GEMMs via V_WMMA_F32_16X16X4_F32 (exact fp32, native matrix pipe).
// - tanh via native V_TANH_F32 trans op (libm tanhf expands to ~25 branchy
//   VALU ops, which would rival the WMMA issue count).
// - Anchor forces reduced algebraically: force = cs*h - sum_k c_k * aN_k.
// - Wave-private 16-row ownership -> no barriers inside the layer loop.

typedef __attribute__((ext_vector_type(2))) float v2f;
typedef __attribute__((ext_vector_type(8))) float v8f;
typedef __attribute__((ext_vector_type(4))) float f4;

#define DIM 256
#define LSTRIDE 260          // 260 % 64 == 4 -> 16 rows spread across all 64 LDS banks
#define ROWS_PER_WG 128
#define ROWS_PER_WAVE 16
#define NLAYERS 6

__device__ __forceinline__ float fast_tanh(float x) {
#if __has_builtin(__builtin_amdgcn_tanhf)
    return __builtin_amdgcn_tanhf(x);
#else
    float y;
    // V_TANH_F32 is a TRANS32 op; ISA requires 1 independent op / V_NOP after a
    // trans op before its output is consumed -> include the v_nop ourselves
    // since the compiler can't see through the asm.
    asm volatile("v_tanh_f32 %0, %1\n\tv_nop" : "=v"(y) : "v"(x));
    return y;
#endif
}

__launch_bounds__(256, 1)
__global__ void collapse_engine_kernel(const float* __restrict__ h0,
                                       const float* __restrict__ W1,
                                       const float* __restrict__ b1,
                                       const float* __restrict__ W2,
                                       const float* __restrict__ b2,
                                       const float* __restrict__ anchors,
                                       float* __restrict__ out,
                                       int Btot) {
    __shared__ float hT[ROWS_PER_WG * LSTRIDE];   // state tile, 133,120 B
    __shared__ float t1T[ROWS_PER_WG * LSTRIDE];  // tanh() activations, 133,120 B
    __shared__ float aN[3 * LSTRIDE];             // normalized anchors
    __shared__ float b1s[DIM];
    __shared__ float b2s[DIM];
    __shared__ float rowCoef[ROWS_PER_WG * 4];    // {c0,c1,c2,cs} per row

    const int tid  = threadIdx.x;
    const int lane = tid & 31;
    const int wid  = tid >> 5;
    const int lm   = lane & 15;   // WMMA N / M lane index
    const int hf   = lane >> 4;   // WMMA K-half / M-half select
    const int blockRow0 = blockIdx.x * ROWS_PER_WG;

    // hint the weight matrices toward cache (global_prefetch_b8)
    __builtin_prefetch(W1 + tid * DIM, 0, 1);
    __builtin_prefetch(W2 + tid * DIM, 0, 1);

    // ---- init: biases ----
    b1s[tid] = b1[tid];
    b2s[tid] = b2[tid];

    // ---- init: normalize anchors (waves 0..2, one anchor each) ----
    if (wid < 3) {
        float s = 0.0f;
        float v[8];
#pragma unroll
        for (int i = 0; i < 8; ++i) {
            float x = anchors[wid * DIM + lane + 32 * i];
            v[i] = x;
            s += x * x;
        }
#pragma unroll
        for (int m = 1; m < 32; m <<= 1) s += __shfl_xor(s, m, 32);
        float inv = 1.0f / fmaxf(sqrtf(s), 1e-12f);
#pragma unroll
        for (int i = 0; i < 8; ++i) aN[wid * LSTRIDE + lane + 32 * i] = v[i] * inv;
    }

    // ---- init: load h tile (float4, coalesced) ----
    {
        const f4* src = (const f4*)(h0 + (size_t)blockRow0 * DIM);
        for (int idx = tid; idx < ROWS_PER_WG * (DIM / 4); idx += 256) {
            int r = idx >> 6, c4 = idx & 63;  // 64 float4 per row
            *(f4*)&hT[r * LSTRIDE + c4 * 4] = src[idx];
        }
    }
    __syncthreads();  // the ONLY barrier: everything below is wave-private

    const size_t alignsOff = (size_t)Btot * DIM;
    const size_t traceSz   = (size_t)NLAYERS * Btot * 3;

    const int rowL = wid * ROWS_PER_WAVE + lm;  // this lane's reduction row (LDS)
    const int bRow = blockRow0 + rowL;          // global row index

    for (int l = 0; l < NLAYERS; ++l) {
        // ================= Phase A: norms, traces, force coefficients ==========
        {
            float hh = 0.f, d0 = 0.f, d1 = 0.f, d2 = 0.f;
            const int cbase = hf * 128;
            const float* hr = &hT[rowL * LSTRIDE];
#pragma unroll 4
            for (int i = 0; i < 128; ++i) {
                int c = cbase + i;
                float v = hr[c];
                hh += v * v;
                d0 += v * aN[0 * LSTRIDE + c];
                d1 += v * aN[1 * LSTRIDE + c];
                d2 += v * aN[2 * LSTRIDE + c];
            }
            hh += __shfl_xor(hh, 16, 32);
            d0 += __shfl_xor(d0, 16, 32);
            d1 += __shfl_xor(d1, 16, 32);
            d2 += __shfl_xor(d2, 16, 32);

            float invn = 1.0f / fmaxf(sqrtf(hh), 1e-12f);
            float al0 = d0 * invn, al1 = d1 * invn, al2 = d2 * invn;
            float dv0 = 1.f - al0, dv1 = 1.f - al1, dv2 = 1.f - al2;
            // ||h - aN_k|| from ||h||^2 - 2 h.aN_k + 1 (anchors normalized)
            float q0 = sqrtf(fmaxf(hh - 2.f * d0 + 1.f, 0.f));
            float q1 = sqrtf(fmaxf(hh - 2.f * d1 + 1.f, 0.f));
            float q2 = sqrtf(fmaxf(hh - 2.f * d2 + 1.f, 0.f));
            float c0 = -0.10f * dv0 / fmaxf(q0, 1e-12f);
            float c1 = -0.10f * dv1 / fmaxf(q1, 1e-12f);
            float c2 = -0.05f * dv2 / fmaxf(q2, 1e-12f);

            if (hf == 0) {
                size_t tb = ((size_t)l * Btot + bRow) * 3;
                out[alignsOff + tb + 0] = al0;
                out[alignsOff + tb + 1] = al1;
                out[alignsOff + tb + 2] = al2;
                out[alignsOff + traceSz + tb + 0] = dv0;
                out[alignsOff + traceSz + tb + 1] = dv1;
                out[alignsOff + traceSz + tb + 2] = dv2;
                out[alignsOff + 2 * traceSz + tb + 0] = dv0 * dv0;
                out[alignsOff + 2 * traceSz + tb + 1] = dv1 * dv1;
                out[alignsOff + 2 * traceSz + tb + 2] = dv2 * dv2;
                rowCoef[rowL * 4 + 0] = c0;
                rowCoef[rowL * 4 + 1] = c1;
                rowCoef[rowL * 4 + 2] = c2;
                rowCoef[rowL * 4 + 3] = c0 + c1 + c2;
            }
        }

        // ================= Phase B: t1 = tanh(h @ W1^T + b1) via WMMA ==========
        {
            const int aRow0 = (wid * ROWS_PER_WAVE + lm) * LSTRIDE + 2 * hf;
            for (int nt = 0; nt < 16; ++nt) {
                v8f acc = {};
                const float* pB = W1 + (nt * 16 + lm) * DIM + 2 * hf;
#pragma unroll 8
                for (int kk = 0; kk < 64; ++kk) {
                    v2f a  = *(const v2f*)&hT[aRow0 + kk * 4];   // ds_load_b64
                    v2f bf = *(const v2f*)&pB[kk * 4];           // global_load_b64
                    acc = __builtin_amdgcn_wmma_f32_16x16x4_f32(
                        false, a, false, bf, (short)0, acc, false, false);
                }
                int col = nt * 16 + lm;
                float bb = b1s[col];
#pragma unroll
                for (int v = 0; v < 8; ++v) {
                    int ldsRow = wid * ROWS_PER_WAVE + v + 8 * hf;  // D-matrix M layout
                    t1T[ldsRow * LSTRIDE + col] = fast_tanh(acc[v] + bb);
                }
            }
        }

        // ===== Phase C: h = h + (t1 @ W2^T + b2) + cs*h - sum_k c_k*aN_k =======
        {
            const int aRow0 = (wid * ROWS_PER_WAVE + lm) * LSTRIDE + 2 * hf;
            float c0v[8], c1v[8], c2v[8], csv[8];
#pragma unroll
            for (int v = 0; v < 8; ++v) {
                int m = wid * ROWS_PER_WAVE + v + 8 * hf;
                c0v[v] = rowCoef[m * 4 + 0];
                c1v[v] = rowCoef[m * 4 + 1];
                c2v[v] = rowCoef[m * 4 + 2];
                csv[v] = rowCoef[m * 4 + 3];
            }
            for (int nt = 0; nt < 16; ++nt) {
                v8f acc = {};
                const float* pB = W2 + (nt * 16 + lm) * DIM + 2 * hf;
#pragma unroll 8
                for (int kk = 0; kk < 64; ++kk) {
                    v2f a  = *(const v2f*)&t1T[aRow0 + kk * 4];
                    v2f bf = *(const v2f*)&pB[kk * 4];
                    acc = __builtin_amdgcn_wmma_f32_16x16x4_f32(
                        false, a, false, bf, (short)0, acc, false, false);
                }
                int col = nt * 16 + lm;
                float bb  = b2s[col];
                float a0c = aN[col];
                float a1c = aN[LSTRIDE + col];
                float a2c = aN[2 * LSTRIDE + col];
#pragma unroll
                for (int v = 0; v < 8; ++v) {
                    int ldsRow = wid * ROWS_PER_WAVE + v + 8 * hf;
                    int idx = ldsRow * LSTRIDE + col;
                    float hold = hT[idx];
                    float val = acc[v] + bb + hold + csv[v] * hold
                              - (c0v[v] * a0c + c1v[v] * a1c + c2v[v] * a2c);
                    hT[idx] = val;  // in-place: this cell not needed elsewhere
                }
            }
        }

        // ================= Phase D: norm clip to MAX_NORM=10 ===================
        {
            float hh = 0.f;
            const int cbase = hf * 128;
            float* hr = &hT[rowL * LSTRIDE];
#pragma unroll 4
            for (int i = 0; i < 128; ++i) {
                float v = hr[cbase + i];
                hh += v * v;
            }
            hh += __shfl_xor(hh, 16, 32);
            float n = sqrtf(hh);
            if (n > 10.0f) {
                float scale = 10.0f / (n + 1e-8f);
#pragma unroll 4
                for (int i = 0; i < 128; ++i) hr[cbase + i] *= scale;
            }
        }
    }

    // ---- write h_final (each wave streams out its own 16 rows) ----
    {
        const float* hrBase = &hT[wid * ROWS_PER_WAVE * LSTRIDE];
        float* outBase = out + (size_t)(blockRow0 + wid * ROWS_PER_WAVE) * DIM;
        for (int idx = lane; idx < ROWS_PER_WAVE * DIM; idx += 32) {
            int r = idx >> 8, c = idx & 255;
            outBase[r * DIM + c] = hrBase[r * LSTRIDE + c];
        }
    }
}

extern "C" void kernel_launch(void* const* d_in, const int* in_sizes, int n_in,
                              void* d_out, int out_size, void* d_ws, size_t ws_size,
                              hipStream_t stream) {
    const float* h0      = (const float*)d_in[0];
    const float* W1      = (const float*)d_in[1];
    const float* b1      = (const float*)d_in[2];
    const float* W2      = (const float*)d_in[3];
    const float* b2      = (const float*)d_in[4];
    const float* anchors = (const float*)d_in[5];
    float* out = (float*)d_out;

    const int Btot = in_sizes[0] / DIM;           // 131072
    const int grid = Btot / ROWS_PER_WG;          // 1024 workgroups

    collapse_engine_kernel<<<dim3(grid), dim3(256), 0, stream>>>(
        h0, W1, b1, W2, b2, anchors, out, Btot);
}